// DilatedReparamConv_76510547411172
// MI455X (gfx1250) — compile-verified
//
#include <hip/hip_runtime.h>
#include <math.h>

#define CCH 256
#define HH 112
#define WW 112
#define NBATCH 8
#define NHW (NBATCH*HH*WW)
#define EPSBN 1e-5f

typedef __attribute__((ext_vector_type(2))) float v2f;
typedef __attribute__((ext_vector_type(4))) float v4f;
typedef __attribute__((ext_vector_type(8))) float v8f;

// ---------------------------------------------------------------------------
// Banded-Toeplitz sheet build: sheet[r][m][kt] (stride 28) holds the 16x28
// Toeplitz matrix of branch row r; x-taps of a (K,D) branch sit at band
// offsets t = 5 + D*(jc - ctr), everything else is zero.
// ---------------------------------------------------------------------------
template<int K, int D>
__device__ __forceinline__ void build_sheet(float* __restrict__ sh,
                                            const float* __restrict__ w,
                                            int c, int tid, int nthreads) {
  constexpr int ctr = (K - 1) / 2;
  for (int idx = tid; idx < K * 448; idx += nthreads) {
    const int r = idx / 448;
    const int m = (idx / 28) & 15;
    const int kt = idx % 28;
    const int t = kt - m;               // band offset in [0,10]
    float v = 0.f;
    const int num = t - 5 + D * ctr;    // = D * jc
    if (t >= 0 && t <= 10 && num >= 0 && (num % D) == 0) {
      const int jc = num / D;
      if (jc < K) v = w[(c * K + r) * K + jc];
    }
    sh[idx] = v;
  }
}

// One branch's Toeplitz WMMA chain: ROWS sheet rows at j = J0 + JSTEP*r,
// CHN K-chunks of 4 along x (chunks past the band are compile-time skipped).
template<int ROWS, int J0, int JSTEP, int CHN>
__device__ __forceinline__ v8f toeplitz_mm(const float* __restrict__ sh,
                                           const float* __restrict__ xw,
                                           int lm, int hi, v8f acc) {
#pragma unroll
  for (int r = 0; r < ROWS; ++r) {
    const int j = J0 + JSTEP * r;
#pragma unroll
    for (int ch = 0; ch < CHN; ++ch) {
      const int kb = 4 * ch + 2 * hi;
      // A (16x4): lane = m + 16*(k>=2), vgprs = k&1
      v2f a = *(const v2f*)&sh[(r * 16 + lm) * 28 + kb];
      // B (4x16): lane = n + 16*(k>=2), vgprs = k&1
      v2f b = *(const v2f*)&xw[(lm + j) * 28 + kb];
      acc = __builtin_amdgcn_wmma_f32_16x16x4_f32(
          false, a, false, b, (short)0, acc, false, false);
    }
  }
  return acc;
}

// ---------------------------------------------------------------------------
// Pass 1: per-branch conv statistics via WMMA. Block = (channel, x-strip);
// the 4 waves split the 6 branches over one shared input tile.
//   wave0: origin(11x11 d1)   wave1: k5_1 + k3_3
//   wave2: k7_1               wave3: k5_2 + k3_5
// Sheet float offsets: 0, 4928, 7168, 10304, 12544, 13888 (total 15232)
// ---------------------------------------------------------------------------
__global__ __launch_bounds__(128) void drb_stats_wmma_kernel(
    const float* __restrict__ x,
    const float* __restrict__ w0, const float* __restrict__ w1,
    const float* __restrict__ w2, const float* __restrict__ w3,
    const float* __restrict__ w4, const float* __restrict__ w5,
    float* __restrict__ part) {
  __shared__ float sheets[15232];   // 60,928 B
  __shared__ float xw[26 * 28];     //  2,912 B (shared by all 4 waves)
  __shared__ float red[128];        //    512 B

  const int bx = blockIdx.x;
  const int c = bx / 7;
  const int tx = bx % 7;
  const int x0 = tx * 16;
  const int tid = threadIdx.x;
  const int lane = tid & 31;
  const int hi = lane >> 4;
  const int lm = lane & 15;
  // scalarized wave id -> scalar branch around the WMMA regions (EXEC all-1s)
  const int wgrp = __builtin_amdgcn_readfirstlane(tid >> 5);

  build_sheet<11, 1>(sheets + 0,     w0, c, tid, 128);
  build_sheet<5, 1>(sheets + 4928,  w1, c, tid, 128);
  build_sheet<7, 1>(sheets + 7168,  w2, c, tid, 128);
  build_sheet<5, 2>(sheets + 10304, w3, c, tid, 128);
  build_sheet<3, 3>(sheets + 12544, w4, c, tid, 128);
  build_sheet<3, 5>(sheets + 13888, w5, c, tid, 128);

  float s1a = 0.f, s2a = 0.f, s1b = 0.f, s2b = 0.f;

  for (int t = 0; t < 56; ++t) {        // 8 batches x 7 y-tiles
    const int n = t / 7;
    const int ty = t % 7;
    const int y0 = ty * 16;
    const float* xp = x + (size_t)(n * CCH + c) * (HH * WW);

    __syncthreads();
    // Stage the shared 26x28 zero-padded tile cooperatively.
    for (int idx = tid; idx < 26 * 28; idx += 128) {
      const int r = idx / 28, cc = idx % 28;
      const int gy = y0 - 5 + r, gx = x0 - 5 + cc;
      float v = 0.f;
      if (gy >= 0 && gy < HH && gx >= 0 && gx < WW) v = xp[gy * WW + gx];
      xw[idx] = v;
    }
    __syncthreads();

    if (t + 1 < 56) {  // prefetch next tile (CDNA5 global_prefetch path)
      const int n2 = (t + 1) / 7, ty2 = (t + 1) % 7;
      __builtin_prefetch(x + (size_t)(n2 * CCH + c) * (HH * WW) +
                             (size_t)(ty2 * 16) * WW + x0, 0, 0);
    }

    v8f accA = {}, accB = {};
    if (wgrp == 0) {
      accA = toeplitz_mm<11, 0, 1, 7>(sheets + 0, xw, lm, hi, accA);
    } else if (wgrp == 1) {
      accA = toeplitz_mm<5, 3, 1, 6>(sheets + 4928, xw, lm, hi, accA);
      accB = toeplitz_mm<3, 2, 3, 6>(sheets + 12544, xw, lm, hi, accB);
    } else if (wgrp == 2) {
      accA = toeplitz_mm<7, 2, 1, 6>(sheets + 7168, xw, lm, hi, accA);
    } else {
      accA = toeplitz_mm<5, 1, 2, 7>(sheets + 10304, xw, lm, hi, accA);
      accB = toeplitz_mm<3, 0, 5, 7>(sheets + 13888, xw, lm, hi, accB);
    }
#pragma unroll
    for (int q = 0; q < 8; ++q) {
      const float ya = accA[q], yb = accB[q];
      s1a += ya; s2a = fmaf(ya, ya, s2a);
      s1b += yb; s2b = fmaf(yb, yb, s2b);   // zeros for waves 0/2
    }
  }

  // slot mapping: wave -> (branchA, branchB)
  const int slotA = (wgrp == 0) ? 0 : (wgrp == 1) ? 1 : (wgrp == 2) ? 2 : 3;
  const int slotB = (wgrp == 1) ? 4 : (wgrp == 3) ? 5 : -1;

  for (int k = 0; k < 12; ++k) {
    float v = 0.f;
    if (k == slotA) v = s1a;
    else if (k == slotA + 6) v = s2a;
    else if (slotB >= 0 && k == slotB) v = s1b;
    else if (slotB >= 0 && k == slotB + 6) v = s2b;
    __syncthreads();
    red[tid] = v;
    __syncthreads();
    for (int s = 64; s > 0; s >>= 1) {
      if (tid < s) red[tid] += red[tid + s];
      __syncthreads();
    }
    if (tid == 0) part[bx * 12 + k] = red[0];
  }
}

// ---------------------------------------------------------------------------
// Pass 2: fold BN into per-branch affine, merge all branches into one 11x11
// depthwise kernel + bias (all branch tap grids embed in the +-5 offset grid)
// ---------------------------------------------------------------------------
template<int K, int D>
__device__ __forceinline__ void embed_kernel(float* __restrict__ Wc,
                                             const float* __restrict__ w,
                                             float a) {
  constexpr int ctr = (K - 1) / 2;
#pragma unroll
  for (int i = 0; i < K; ++i)
#pragma unroll
    for (int j = 0; j < K; ++j)
      Wc[(5 + D * (i - ctr)) * 11 + (5 + D * (j - ctr))] =
          fmaf(a, w[i * K + j], Wc[(5 + D * (i - ctr)) * 11 + (5 + D * (j - ctr))]);
}

__global__ __launch_bounds__(256) void drb_merge_kernel(
    const float* __restrict__ part,
    const float* __restrict__ w0, const float* __restrict__ g0, const float* __restrict__ be0,
    const float* __restrict__ w1, const float* __restrict__ g1, const float* __restrict__ be1,
    const float* __restrict__ w2, const float* __restrict__ g2, const float* __restrict__ be2,
    const float* __restrict__ w3, const float* __restrict__ g3, const float* __restrict__ be3,
    const float* __restrict__ w4, const float* __restrict__ g4, const float* __restrict__ be4,
    const float* __restrict__ w5, const float* __restrict__ g5, const float* __restrict__ be5,
    float* __restrict__ Wm, float* __restrict__ bias) {
  const int c = threadIdx.x;
  const float* gs[6] = {g0, g1, g2, g3, g4, g5};
  const float* bs[6] = {be0, be1, be2, be3, be4, be5};

  float a[6];
  float cb = 0.f;
  const float inv = 1.f / (float)NHW;
#pragma unroll
  for (int b = 0; b < 6; ++b) {
    float sy = 0.f, syy = 0.f;
    for (int q = 0; q < 7; ++q) {       // 7 x-strip partials per channel
      sy  += part[(c * 7 + q) * 12 + b];
      syy += part[(c * 7 + q) * 12 + 6 + b];
    }
    const float mean = sy * inv;
    const float var  = syy * inv - mean * mean;
    const float ab   = gs[b][c] * rsqrtf(var + EPSBN);
    a[b] = ab;
    cb += bs[b][c] - ab * mean;
  }

  float* Wc = Wm + c * 121;
  for (int t = 0; t < 121; ++t) Wc[t] = a[0] * w0[c * 121 + t];
  embed_kernel<5, 1>(Wc, w1 + c * 25, a[1]);
  embed_kernel<7, 1>(Wc, w2 + c * 49, a[2]);
  embed_kernel<5, 2>(Wc, w3 + c * 25, a[3]);
  embed_kernel<3, 3>(Wc, w4 + c * 9, a[4]);
  embed_kernel<3, 5>(Wc, w5 + c * 9, a[5]);
  bias[c] = cb;
}

// ---------------------------------------------------------------------------
// Pass 3: merged 11x11 depthwise conv via banded-Toeplitz WMMA (f32 16x16x4)
// ---------------------------------------------------------------------------
__global__ __launch_bounds__(128) void drb_conv_wmma_kernel(
    const float* __restrict__ x, const float* __restrict__ Wm,
    const float* __restrict__ bias, float* __restrict__ out) {
  __shared__ float T[11 * 16 * 28];    // Toeplitz sheets, stride 28 (bank-clean)
  __shared__ float xt[4][26 * 28];     // per-wave input tiles

  const int bx = blockIdx.x;
  const int c = bx / 7;
  const int tx = bx % 7;
  const int x0 = tx * 16;
  const int tid = threadIdx.x;
  const int wave = __builtin_amdgcn_readfirstlane(tid >> 5);
  const int lane = tid & 31;
  const int hi = lane >> 4;
  const int lm = lane & 15;

  // T[j][m][kt] = W[j][kt-m] (dense 11-band: merged kernel)
  const float* Wc = Wm + c * 121;
  for (int idx = tid; idx < 11 * 16 * 28; idx += 128) {
    const int j = idx / 448;
    const int m = (idx / 28) & 15;
    const int kt = idx % 28;
    const int t = kt - m;
    T[idx] = (t >= 0 && t <= 10) ? Wc[j * 11 + t] : 0.f;
  }
  const float bc = bias[c];
  __syncthreads();

  float* xw = xt[wave];
  for (int t = wave; t < 56; t += 4) {      // 8 batches x 7 y-tiles
    const int n = t / 7;
    const int ty = t % 7;
    const int y0 = ty * 16;
    const float* xp = x + (size_t)(n * CCH + c) * (HH * WW);

    for (int idx = lane; idx < 26 * 28; idx += 32) {
      const int r = idx / 28, cc = idx % 28;
      const int gy = y0 - 5 + r, gx = x0 - 5 + cc;
      float v = 0.f;
      if (gy >= 0 && gy < HH && gx >= 0 && gx < WW) v = xp[gy * WW + gx];
      xw[idx] = v;
    }
    __builtin_amdgcn_wave_barrier();  // same-wave DS ops complete in order

    if (t + 4 < 56) {
      const int n2 = (t + 4) / 7, ty2 = (t + 4) % 7;
      __builtin_prefetch(x + (size_t)(n2 * CCH + c) * (HH * WW) +
                             (size_t)(ty2 * 16) * WW + x0, 0, 0);
    }

    v8f acc = {};
    acc = toeplitz_mm<11, 0, 1, 7>(T, xw, lm, hi, acc);
#pragma unroll
    for (int q = 0; q < 8; ++q) acc[q] += bc;

    // D: lane holds row y0+lm, 8 consecutive x at x0+8*hi -> 32B contiguous
    float* op = out + (size_t)(n * CCH + c) * (HH * WW) +
                (size_t)(y0 + lm) * WW + x0 + 8 * hi;
    *(v4f*)op       = v4f{acc[0], acc[1], acc[2], acc[3]};
    *(v4f*)(op + 4) = v4f{acc[4], acc[5], acc[6], acc[7]};
    __builtin_amdgcn_wave_barrier();
  }
}

// ---------------------------------------------------------------------------
extern "C" void kernel_launch(void* const* d_in, const int* in_sizes, int n_in,
                              void* d_out, int out_size, void* d_ws, size_t ws_size,
                              hipStream_t stream) {
  const float* x  = (const float*)d_in[0];
  const float* w0 = (const float*)d_in[1];
  const float* g0 = (const float*)d_in[2];
  const float* b0 = (const float*)d_in[3];
  const float* w1 = (const float*)d_in[4];
  const float* g1 = (const float*)d_in[5];
  const float* b1 = (const float*)d_in[6];
  const float* w2 = (const float*)d_in[7];
  const float* g2 = (const float*)d_in[8];
  const float* b2 = (const float*)d_in[9];
  const float* w3 = (const float*)d_in[10];
  const float* g3 = (const float*)d_in[11];
  const float* b3 = (const float*)d_in[12];
  const float* w4 = (const float*)d_in[13];
  const float* g4 = (const float*)d_in[14];
  const float* b4 = (const float*)d_in[15];
  const float* w5 = (const float*)d_in[16];
  const float* g5 = (const float*)d_in[17];
  const float* b5 = (const float*)d_in[18];
  float* out = (float*)d_out;

  float* ws   = (float*)d_ws;
  float* part = ws;                    // 1792 * 12 floats
  float* Wm   = ws + 1792 * 12;        // 256 * 121 floats
  float* bias = Wm + 256 * 121;        // 256 floats

  drb_stats_wmma_kernel<<<256 * 7, 128, 0, stream>>>(x, w0, w1, w2, w3, w4, w5,
                                                     part);
  drb_merge_kernel<<<1, 256, 0, stream>>>(part,
                                          w0, g0, b0, w1, g1, b1, w2, g2, b2,
                                          w3, g3, b3, w4, g4, b4, w5, g5, b5,
                                          Wm, bias);
  drb_conv_wmma_kernel<<<256 * 7, 128, 0, stream>>>(x, Wm, bias, out);
}